// GAMReal_Module_24833500905654
// MI455X (gfx1250) — compile-verified
//
#include <hip/hip_runtime.h>
#include <hip/hip_bf16.h>

typedef float v2f __attribute__((ext_vector_type(2)));
typedef float v8f __attribute__((ext_vector_type(8)));

#define NB 16
#define NG 8
#define NN 131072            // (C/G)*H*W = 32*4096
#define KT 256               // K-tile staged in LDS per step
#define LDS_STRIDE 260       // padded row stride (floats)
#define TILE_F (16 * LDS_STRIDE)
#define GR_SLICES 64         // N-slices per batch-pair for gram pass
#define NS (NN / GR_SLICES)  // 2048 columns per gram block
#define MIX_SLICES 32
#define MIX_COLS (NN / MIX_SLICES) // 4096 columns per mix block

// ---------------- pass 0: zero gram accumulator ----------------
__global__ void zero_gram(float* __restrict__ g) {
    int i = blockIdx.x * blockDim.x + threadIdx.x;
    if (i < NB * NG * NG) g[i] = 0.0f;
}

// ---------------- pass 1: grouped gram via WMMA ----------------
// grid = (8 batch-pairs, 64 slices), block = 256 (8 waves).
// Each block computes a partial 16x16 gram tile (2 batches x 8 groups)
// over NS columns and atomically accumulates the two diagonal 8x8 blocks.
// LDS double-buffered: one barrier per K-tile; staging loads issued before
// the WMMA chain so HBM traffic overlaps matrix compute.
__global__ __launch_bounds__(256) void gram_kernel(const float* __restrict__ x,
                                                   float* __restrict__ gram) {
    __shared__ __align__(16) float tile[2][TILE_F];
    __shared__ float red[8][256];

    const int bp   = blockIdx.x;          // batch pair: batches 2bp, 2bp+1
    const int n0   = blockIdx.y * NS;     // column slice start
    const int tid  = threadIdx.x;
    const int wave = tid >> 5;
    const int lane = tid & 31;

    // rows (b,g) for this pair are contiguous: global row = 16*bp + r
    const float* base = x + (size_t)(16 * bp) * NN + n0;

    // per-thread staging coordinates (constant across tiles)
    int s_row[4], s_c4[4];
    #pragma unroll
    for (int j = 0; j < 4; ++j) {
        int f    = tid + 256 * j;         // float4 index 0..1023
        s_row[j] = f >> 6;                // 64 float4 per row
        s_c4[j]  = (f & 63) << 2;         // column (floats)
    }

    // A-fragment coords for V_WMMA_F32_16X16X4_F32:
    //   lane = M (mod 16); VGPR0 holds K = k0 + (lane<16 ? 0 : 2), VGPR1 = +1.
    // B (4x16) uses the same lane/K split with lane = N, so for a gram
    // (B == A^T of the same data tile) we pass identical registers.
    const int m     = lane & 15;
    const int khalf = (lane >> 4) << 1;   // 0 or 2

    v8f acc = {};                         // 16x16 f32 accumulator (8 VGPRs)

    // prologue: stage tile 0 into buffer 0
    {
        float4 st[4];
        #pragma unroll
        for (int j = 0; j < 4; ++j)
            st[j] = *(const float4*)(base + (size_t)s_row[j] * NN + s_c4[j]);
        #pragma unroll
        for (int j = 0; j < 4; ++j)
            *(float4*)&tile[0][s_row[j] * LDS_STRIDE + s_c4[j]] = st[j];
    }
    __syncthreads();

    int p = 0;
    for (int t = 0; t < NS; t += KT) {
        const bool more = (t + KT < NS);
        float4 st[4];
        if (more) {
            // issue next tile's loads up front; they complete under the WMMAs
            #pragma unroll
            for (int j = 0; j < 4; ++j)
                st[j] = *(const float4*)(base + (size_t)s_row[j] * NN + (t + KT) + s_c4[j]);
            if (t + 2 * KT < NS) {
                #pragma unroll
                for (int j = 0; j < 4; ++j)
                    __builtin_prefetch(base + (size_t)s_row[j] * NN + (t + 2 * KT) + s_c4[j], 0, 1);
            }
        }

        // this wave's K-subrange [wave*32, wave*32+32) of the current tile
        const float* tp = &tile[p][m * LDS_STRIDE + (wave << 5) + khalf];
        #pragma unroll
        for (int kk = 0; kk < 32; kk += 4) {
            v2f a = *(const v2f*)(tp + kk);
            acc = __builtin_amdgcn_wmma_f32_16x16x4_f32(
                      false, a, false, a, (short)0, acc, false, false);
        }

        if (more) {
            #pragma unroll
            for (int j = 0; j < 4; ++j)
                *(float4*)&tile[p ^ 1][s_row[j] * LDS_STRIDE + s_c4[j]] = st[j];
        }
        __syncthreads();
        p ^= 1;
    }

    // cross-wave reduction of the 16x16 accumulator
    #pragma unroll
    for (int i = 0; i < 8; ++i) red[wave][i * 32 + lane] = acc[i];
    __syncthreads();

    {
        // element tid: VGPR r = tid/32, lane l = tid%32
        const int r = tid >> 5, l = tid & 31;
        float s = 0.0f;
        #pragma unroll
        for (int w = 0; w < 8; ++w) s += red[w][tid];
        // D layout: m = r + 8*(l>=16), n = l%16
        const int mm = r + ((l >= 16) ? 8 : 0);
        const int nn = l & 15;
        if ((mm < 8) == (nn < 8)) {   // keep only per-batch diagonal blocks
            int b = 2 * bp + (mm >= 8);
            atomicAdd(&gram[b * 64 + (mm & 7) * 8 + (nn & 7)], s);
        }
    }
}

// ---------------- pass 2: softmax + group mixing + residual ----------------
// grid = (16 batches, 32 slices), block = 256.
__global__ __launch_bounds__(256) void mix_kernel(const float* __restrict__ x,
                                                  const float* __restrict__ gram,
                                                  const float* __restrict__ gamma,
                                                  float* __restrict__ out) {
    __shared__ float eg[64];
    __shared__ float ex[64];
    __shared__ float attn[64];

    const int b   = blockIdx.x;
    const int tid = threadIdx.x;

    if (tid < 64) eg[tid] = gram[b * 64 + tid];
    __syncthreads();
    if (tid < 64) {
        // softmax(rowmax - e) == softmax(-e); stable shift by rowmin of e
        int g = tid >> 3;
        float mn = eg[g * 8];
        #pragma unroll
        for (int h = 1; h < 8; ++h) mn = fminf(mn, eg[g * 8 + h]);
        ex[tid] = expf(mn - eg[tid]);   // <= 0 exponent, no overflow
    }
    __syncthreads();
    if (tid < 64) {
        int g = tid >> 3;
        float s = 0.0f;
        #pragma unroll
        for (int h = 0; h < 8; ++h) s += ex[g * 8 + h];
        attn[tid] = ex[tid] / s;
    }
    __syncthreads();

    float a[64];
    #pragma unroll
    for (int i = 0; i < 64; ++i) a[i] = attn[i];
    const float gm = gamma[0];

    const float* xb = x   + (size_t)b * NG * NN;
    float*       ob = out + (size_t)b * NG * NN;
    const int base_n = blockIdx.y * MIX_COLS;

    for (int it = 0; it < MIX_COLS / 1024; ++it) {
        const int n4 = base_n + it * 1024 + (tid << 2);
        float4 vv[8];
        #pragma unroll
        for (int h = 0; h < 8; ++h)
            vv[h] = *(const float4*)(xb + (size_t)h * NN + n4);
        #pragma unroll
        for (int g = 0; g < 8; ++g) {
            float sx = 0.f, sy = 0.f, sz = 0.f, sw = 0.f;
            #pragma unroll
            for (int h = 0; h < 8; ++h) {
                const float w = a[g * 8 + h];
                sx = fmaf(w, vv[h].x, sx);
                sy = fmaf(w, vv[h].y, sy);
                sz = fmaf(w, vv[h].z, sz);
                sw = fmaf(w, vv[h].w, sw);
            }
            float4 o;
            o.x = fmaf(gm, sx, vv[g].x);
            o.y = fmaf(gm, sy, vv[g].y);
            o.z = fmaf(gm, sz, vv[g].z);
            o.w = fmaf(gm, sw, vv[g].w);
            *(float4*)(ob + (size_t)g * NN + n4) = o;
        }
    }
}

extern "C" void kernel_launch(void* const* d_in, const int* in_sizes, int n_in,
                              void* d_out, int out_size, void* d_ws, size_t ws_size,
                              hipStream_t stream) {
    const float* x     = (const float*)d_in[0];
    const float* gamma = (const float*)d_in[1];
    // d_in[2] is no_g == 8, hard-coded above
    float* out  = (float*)d_out;
    float* gram = (float*)d_ws;   // 16*8*8 floats = 4 KB scratch

    zero_gram<<<dim3(1), dim3(1024), 0, stream>>>(gram);
    gram_kernel<<<dim3(8, GR_SLICES), dim3(256), 0, stream>>>(x, gram);
    mix_kernel<<<dim3(NB, MIX_SLICES), dim3(256), 0, stream>>>(x, gram, gamma, out);
}